// DenseTableBatchedEmbeddingBagsCodegen_65369402245266
// MI455X (gfx1250) — compile-verified
//
#include <hip/hip_runtime.h>
#include <stdint.h>

// Table-batched EmbeddingBag (SUM pooling), tuned for MI455X / gfx1250.
// HBM-bound gather-reduce (~0.7 GB random 512B-row reads per call -> ~30us
// at 23.3 TB/s). Uses the CDNA5 async global->LDS path
// (GLOBAL_LOAD_ASYNC_TO_LDS_B128 + ASYNCcnt): each wave-instruction gathers
// one full 512B embedding row into LDS with zero destination-VGPR pressure,
// keeping 20 row fetches in flight per wave.

namespace {
constexpr int kT        = 8;       // tables
constexpr int kB        = 8192;    // batch
constexpr int kD        = 128;     // embedding dim
constexpr int kRows     = 200000;  // rows per table
constexpr int kWaves    = 4;       // waves per block (128 threads)
constexpr int kChunk    = 20;      // LDS row slots per wave (L == 20)
constexpr int kHalf     = 10;      // split point for pipelined reduce
constexpr int kRowBytes = kD * 4;  // 512 bytes per embedding row
}

typedef float v4f __attribute__((ext_vector_type(4)));

// One 16B-per-lane async gather: wave pulls a full 512B row into LDS.
// GV mode: 64-bit per-lane global address in a VGPR pair, saddr = off.
// Tracked by ASYNCcnt; completes in order with other async loads.
__device__ __forceinline__ void async_row16(uint32_t lds_addr, const float* gaddr) {
  asm volatile("global_load_async_to_lds_b128 %0, %1, off"
               :
               : "v"(lds_addr), "v"(gaddr)
               : "memory");
}

__device__ __forceinline__ void wait_async_le10() {
  asm volatile("s_wait_asynccnt 0xa" ::: "memory");
}
__device__ __forceinline__ void wait_async0() {
  asm volatile("s_wait_asynccnt 0x0" ::: "memory");
}

__global__ __launch_bounds__(kWaves * 32, 1)
void embedding_bag_async_kernel(const int* __restrict__ indices,
                                const int* __restrict__ offsets,
                                const float* __restrict__ weights,
                                float* __restrict__ out,
                                int nbags) {
  // 4 waves * 20 rows * 512B = 40 KB static LDS -> 8 blocks (32 waves) / WGP.
  __shared__ __align__(16) float smem[kWaves * kChunk * kD];

  const int lane = (int)(threadIdx.x & 31u);
  // Force wave-uniform bag id so offsets/indices loads become s_load.
  const int wid  = __builtin_amdgcn_readfirstlane((int)(threadIdx.x >> 5));
  const int bag  = (int)blockIdx.x * kWaves + wid;
  if (bag >= nbags) return;   // whole-wave exit; EXEC stays all-ones otherwise

  const int t = bag / kB;        // table (feature) index
  const int b = bag - t * kB;    // batch index

  const int start = offsets[bag];
  const int end   = offsets[bag + 1];

  float* swave = smem + wid * (kChunk * kD);
  // Generic LDS pointer truncates to the 32-bit LDS byte offset.
  const uint32_t lds_lane_base =
      (uint32_t)(uintptr_t)swave + (uint32_t)(lane * 16);

  // Per-lane base into this table's rows: lane covers dims [4*lane, 4*lane+4).
  const float* wbase = weights + (size_t)t * (size_t)kRows * kD + (size_t)(lane * 4);

  v4f acc = {0.f, 0.f, 0.f, 0.f};

  for (int s = start; s < end; s += kChunk) {
    int n = end - s;
    if (n > kChunk) n = kChunk;

    if (n == kChunk) {
      // ---- Fast path: L == 20 (the provided inputs). Fully unrolled so the
      // 20 uniform index loads merge into wide s_load_bN with one kmcnt wait,
      // and the 20 async row gathers issue back-to-back (ASYNCcnt max 63).
      const int* __restrict__ ib = indices + s;
#pragma unroll
      for (int r = 0; r < kChunk; ++r) {
        const int row = ib[r];                        // scalar load (uniform)
        async_row16(lds_lane_base + (uint32_t)(r * kRowBytes),
                    wbase + (size_t)row * kD);
      }

      // Async loads complete in order: <=10 outstanding => rows 0..9 landed.
      wait_async_le10();
#pragma unroll
      for (int r = 0; r < kHalf; ++r) {
        acc += *(const v4f*)(swave + r * kD + lane * 4);
      }
      wait_async0();
#pragma unroll
      for (int r = kHalf; r < kChunk; ++r) {
        acc += *(const v4f*)(swave + r * kD + lane * 4);
      }
    } else {
      // ---- Generic ragged tail path.
      for (int r = 0; r < n; ++r) {
        const int row = indices[s + r];
        async_row16(lds_lane_base + (uint32_t)(r * kRowBytes),
                    wbase + (size_t)row * kD);
      }
      wait_async0();
      for (int r = 0; r < n; ++r) {
        acc += *(const v4f*)(swave + r * kD + lane * 4);
      }
    }
    // WAR on LDS across chunks is safe: the adds consume the ds loads
    // (s_wait_dscnt inserted before use), and the next chunk's asm issues
    // only after them in program order.
  }

  // out[b][t*D + lane*4 .. +4)  (batch-major, features concatenated)
  float* op = out + (size_t)b * (size_t)(kT * kD) + (size_t)t * kD + (size_t)(lane * 4);
  *(v4f*)op = acc;
}

extern "C" void kernel_launch(void* const* d_in, const int* in_sizes, int n_in,
                              void* d_out, int out_size, void* d_ws, size_t ws_size,
                              hipStream_t stream) {
  const int*   indices = (const int*)d_in[0];
  const int*   offsets = (const int*)d_in[1];
  const float* weights = (const float*)d_in[2];
  float*       out     = (float*)d_out;

  const int nbags  = in_sizes[1] - 1;                 // T*B = 65536
  const int blocks = (nbags + kWaves - 1) / kWaves;   // 16384

  hipLaunchKernelGGL(embedding_bag_async_kernel,
                     dim3(blocks), dim3(kWaves * 32), 0, stream,
                     indices, offsets, weights, out, nbags);
}